// GlobalRaggedConvolution_45612552683661
// MI455X (gfx1250) — compile-verified
//
#include <hip/hip_runtime.h>
#include <hip/hip_bf16.h>
#include <stdint.h>

// Problem constants (match reference)
#define NPTS   500000
#define FIN    64
#define UNITS  128
#define CDIM   4
#define NSEG   32

#define TILE   64      // points staged per LDS tile
#define PTSB   1024    // points per block (16 tiles)

typedef __attribute__((ext_vector_type(2))) float v2f;
typedef __attribute__((ext_vector_type(8))) float v8f;

// ---- CDNA5 async global->LDS copy of 16 bytes per lane ----
__device__ __forceinline__ void async_cp16(uint32_t lds_off, const float* gptr) {
    asm volatile("global_load_async_to_lds_b128 %0, %1, off"
                 :: "v"(lds_off), "v"(gptr) : "memory");
}
__device__ __forceinline__ void wait_async0() {
    asm volatile("s_wait_asynccnt 0" ::: "memory");
}
__device__ __forceinline__ uint32_t lds_addr32(const void* p) {
    // LDS flat aperture keeps the byte offset in the low 32 bits
    return (uint32_t)(uintptr_t)p;
}

// =====================================================================
// Stage 0: zero the workspace accumulators (harness poisons d_ws)
// =====================================================================
__global__ void zero_ws_kernel(float* p, int n) {
    int i = blockIdx.x * blockDim.x + threadIdx.x;
    if (i < n) p[i] = 0.0f;
}

// =====================================================================
// Stage 1: streaming per-segment outer products
//   G[s][k][c]  += x[n,k]*coord[n,c]   (k<64, c<4)
//   CS[s][c]    += coord[n,c]
// Memory-bound: 128MB of node features streamed once via async-LDS DMA.
// =====================================================================
__global__ __launch_bounds__(256) void stage1_kernel(
        const float* __restrict__ node, const float* __restrict__ coord,
        const int* __restrict__ rsplits, float* __restrict__ G,
        float* __restrict__ CS, int n) {
    __shared__ __align__(16) float s_x[2][TILE * FIN];
    __shared__ __align__(16) float s_c[2][TILE * CDIM];
    __shared__ int s_rs[NSEG + 1];

    const int t = threadIdx.x;
    const int start = blockIdx.x * PTSB;
    if (start >= n) return;
    const int end = min(n, start + PTSB);

    if (t < NSEG + 1) s_rs[t] = rsplits[t];
    __syncthreads();

    const int k = t >> 2;       // feature index owned by this thread
    const int c = t & 3;        // coord index owned by this thread

    // segment of first point: seg = max i with rs[i] <= start (upper_bound-1)
    int seg = 0;
    while (s_rs[seg + 1] <= start) ++seg;
    int seg_end = s_rs[seg + 1];

    float acc = 0.0f;   // sum x[n,k]*coord[n,c] over current segment
    float cs  = 0.0f;   // sum coord[n,c]       over current segment

    const int ntile = (end - start + TILE - 1) / TILE;

    // --- preload tile 0 ---
    {
        const int tn = min(TILE, end - start);
        const float* gb = node + (size_t)start * FIN;
        const uint32_t lb = lds_addr32(&s_x[0][0]);
        const int nchunk = tn * (FIN / 4);
        for (int j = t; j < TILE * (FIN / 4); j += 256)
            if (j < nchunk) async_cp16(lb + (uint32_t)j * 16u, gb + j * 4);
        const float* gc = coord + (size_t)start * CDIM;
        const uint32_t lc = lds_addr32(&s_c[0][0]);
        if (t < tn) async_cp16(lc + (uint32_t)t * 16u, gc + t * 4);
    }

    for (int ti = 0; ti < ntile; ++ti) {
        const int buf = ti & 1;
        const int tstart = start + ti * TILE;
        const int tn = min(TILE, end - tstart);

        wait_async0();          // this wave's tile-ti DMA done
        __syncthreads();        // everyone's DMA done -> tile readable

        // prefetch tile ti+1 into the other buffer (overlaps compute)
        if (ti + 1 < ntile) {
            const int ns = tstart + TILE;
            const int nn = min(TILE, end - ns);
            const float* gb = node + (size_t)ns * FIN;
            const uint32_t lb = lds_addr32(&s_x[buf ^ 1][0]);
            const int nchunk = nn * (FIN / 4);
            for (int j = t; j < TILE * (FIN / 4); j += 256)
                if (j < nchunk) async_cp16(lb + (uint32_t)j * 16u, gb + j * 4);
            const float* gc = coord + (size_t)ns * CDIM;
            const uint32_t lc = lds_addr32(&s_c[buf ^ 1][0]);
            if (t < nn) async_cp16(lc + (uint32_t)t * 16u, gc + t * 4);
        }

        // process tile ti, splitting at segment boundaries (uniform scalars)
        int pl = 0;
        while (pl < tn) {
            int run = seg_end - tstart;      // >= pl always; may be > tn
            if (run > tn) run = tn;
            #pragma unroll 4
            for (; pl < run; ++pl) {
                const float xv = s_x[buf][pl * FIN + k];   // broadcast, no conflicts
                const float cv = s_c[buf][pl * CDIM + c];  // broadcast
                acc = __builtin_fmaf(xv, cv, acc);
                cs += cv;                                  // meaningful for t<4
            }
            if (pl < tn) {                   // crossed a segment boundary
                atomicAdd(&G[seg * (FIN * CDIM) + t], acc);
                acc = 0.0f;
                if (t < CDIM) atomicAdd(&CS[seg * CDIM + t], cs);
                cs = 0.0f;
                const int p = tstart + pl;
                while (s_rs[seg + 1] <= p) ++seg;
                seg_end = s_rs[seg + 1];
            }
        }
    }

    // final flush
    atomicAdd(&G[seg * (FIN * CDIM) + t], acc);
    if (t < CDIM) atomicAdd(&CS[seg * CDIM + t], cs);
}

// =====================================================================
// Stage 2: out[32,128] = Gm[32,256] @ W2[256,128] + CS[32,4] @ b2[4,128]
//   Gm[s, 4k+c] = G[s][k][c];  W2[4k+c, u] = W[k*512 + u*4 + c]
// V_WMMA_F32_16X16X4_F32, one 16x16 tile per wave, K-loop of 64.
// =====================================================================
__global__ __launch_bounds__(32) void stage2_kernel(
        const float* __restrict__ G, const float* __restrict__ CS,
        const float* __restrict__ W, const float* __restrict__ bias,
        float* __restrict__ out) {
    const int wave = blockIdx.x;       // 0..15
    const int mt = wave >> 3;          // row tile (s)
    const int nt = wave & 7;           // col tile (u)
    const int lane = threadIdx.x;      // 0..31
    const int l15 = lane & 15;
    const int lhi = lane >> 4;         // 0 or 1

    const int srow = mt * 16 + l15;    // A-frag M row
    const int ucol = nt * 16 + l15;    // B-frag N col

    v8f cacc = {};
    for (int kk = 0; kk < (FIN * CDIM) / 4; ++kk) {
        const int k0 = kk * 4 + lhi * 2;     // K index of reg 0
        v2f a, bm;
        // A: Gm[srow][k0], Gm[srow][k0+1]
        a.x = G[srow * (FIN * CDIM) + k0];
        a.y = G[srow * (FIN * CDIM) + k0 + 1];
        // B: W2[k0][ucol], W2[k0+1][ucol]
        bm.x = W[(k0 >> 2) * (UNITS * CDIM) + ucol * CDIM + (k0 & 3)];
        bm.y = W[((k0 + 1) >> 2) * (UNITS * CDIM) + ucol * CDIM + ((k0 + 1) & 3)];
        cacc = __builtin_amdgcn_wmma_f32_16x16x4_f32(
                   false, a, false, bm, (short)0, cacc, false, false);
    }

    // bias term: + sum_c CS[s][c] * b[u*4+c], then store
    const float b0 = bias[ucol * CDIM + 0];
    const float b1 = bias[ucol * CDIM + 1];
    const float b2 = bias[ucol * CDIM + 2];
    const float b3 = bias[ucol * CDIM + 3];
    #pragma unroll
    for (int v = 0; v < 8; ++v) {
        const int s = mt * 16 + v + 8 * lhi;   // C/D layout: M = v + 8*(lane>=16)
        const float add = CS[s * CDIM + 0] * b0 + CS[s * CDIM + 1] * b1 +
                          CS[s * CDIM + 2] * b2 + CS[s * CDIM + 3] * b3;
        out[s * UNITS + nt * 16 + l15] = cacc[v] + add;
    }
}

// =====================================================================
extern "C" void kernel_launch(void* const* d_in, const int* in_sizes, int n_in,
                              void* d_out, int out_size, void* d_ws, size_t ws_size,
                              hipStream_t stream) {
    const float* node  = (const float*)d_in[0];   // [N,64]
    const float* coord = (const float*)d_in[1];   // [N,4]
    const float* W     = (const float*)d_in[2];   // [64,512]
    const float* bias  = (const float*)d_in[3];   // [512]
    const int*   rs    = (const int*)d_in[4];     // [33]
    float* out = (float*)d_out;                   // [32,128]

    float* G  = (float*)d_ws;                     // 32*64*4 = 8192 floats
    float* CS = G + NSEG * FIN * CDIM;            // 32*4    = 128 floats
    const int accn = NSEG * FIN * CDIM + NSEG * CDIM;

    zero_ws_kernel<<<(accn + 255) / 256, 256, 0, stream>>>(G, accn);

    const int nblk = (NPTS + PTSB - 1) / PTSB;    // 489 blocks
    stage1_kernel<<<nblk, 256, 0, stream>>>(node, coord, rs, G, CS, NPTS);

    stage2_kernel<<<16, 32, 0, stream>>>(G, CS, W, bias, out);
}